// Symbolic_KANLayer_34497177321999
// MI455X (gfx1250) — compile-verified
//
#include <hip/hip_runtime.h>
#include <math.h>

// ---------------------------------------------------------------------------
// Symbolic KAN layer for MI455X (gfx1250, wave32).
//
//   x_red = x @ mask^T                 -> V_WMMA_F32_16X16X4_F32 tiles
//   z     = a * x_red + b               per (b, r, i)
//   post  = c * f_{fun_ids[r,i]}(z) + d -> 268 MB streaming store
//   y     = sum_i post
//
// Roofline: 268 MB stores / 23.3 TB/s = 11.6 us floor. fun_ids is constant
// across batch, so the activation kernel keeps (r, i) wave-uniform and puts
// batch on lanes: one scalar branch -> ONE function body per iteration
// instead of 10 divergent bodies. LDS staging keeps all global traffic
// coalesced float4.
// ---------------------------------------------------------------------------

typedef __attribute__((ext_vector_type(2))) float v2f;
typedef __attribute__((ext_vector_type(8))) float v8f;

#define IN_F  128
#define OUT_F 128
#define BT    32      // batch tile per workgroup

// ---------------------------------------------------------------------------
// Kernel 1: x_red[b, r] = sum_i x[b, i] * mask[r, i]
// One wave per 16x16 tile, K=128 -> 32 chained v_wmma_f32_16x16x4_f32.
// A (16x4 f32): lane L holds M=L%16; lanes<16: K={0,1}, lanes>=16: K={2,3}.
// B (4x16 f32): N=L%16, same K split. C/D: row = vgpr + 8*(L>=16), col=L%16.
// ---------------------------------------------------------------------------
__global__ void __launch_bounds__(32)
xred_wmma_kernel(const float* __restrict__ X,     // (batch, IN_F)
                 const float* __restrict__ Mk,    // (IN_F, IN_F) mask
                 float*       __restrict__ Xred)  // (batch, IN_F)
{
    const int lane   = threadIdx.x & 31;
    const int lane16 = lane & 15;
    const int half   = lane >> 4;
    const int m0     = blockIdx.x << 4;
    const int n0     = blockIdx.y << 4;

    const float* arow = X  + (size_t)(m0 + lane16) * IN_F;
    const float* brow = Mk + (size_t)(n0 + lane16) * IN_F;

    v8f acc = {};
#pragma unroll
    for (int k0 = 0; k0 < IN_F; k0 += 4) {
        const int ka = k0 + half * 2;
        v2f afrag = *(const v2f*)(arow + ka);
        v2f bfrag = *(const v2f*)(brow + ka);
        acc = __builtin_amdgcn_wmma_f32_16x16x4_f32(
            false, afrag, false, bfrag, (short)0, acc, false, false);
    }

#pragma unroll
    for (int j = 0; j < 8; ++j) {
        const int row = m0 + j + half * 8;
        Xred[(size_t)row * IN_F + n0 + lane16] = acc[j];
    }
}

// ---------------------------------------------------------------------------
// Function bodies. fid is wave-uniform (SGPR) -> scalar branch picks ONE.
// Native v_sin_f32 / v_exp_f32 for the hot simple cases (z is O(few units));
// tanhf/atanf stay precise (executed only when selected, uniform).
// ---------------------------------------------------------------------------
__device__ __forceinline__ float apply_fun_u(float z, int fid) {
    switch (fid) {
        case 0:  return z;
        case 1:  return z * z;
        case 2:  return z * z * z;
        case 3:  return __sinf(z);                    // v_sin_f32
        case 4:  return tanhf(z);
        case 5:  return 1.0f / (1.0f + __expf(-z));   // v_exp_f32
        case 6:  return __expf(-z * z);
        case 7:  return fabsf(z);
        case 8:  return atanf(z);
        default: return __expf(z);                    // case 9
    }
}

// ---------------------------------------------------------------------------
// Kernel 2: workgroup = (BT batch rows) x (all 128 i) x (one r).
// Wave w covers i in [16w, 16w+16): i, fid, a/b/c/d all wave-uniform
// (scalar loads + scalar branch); lanes carry 32 batch rows.
// LDS tiles padded to IN_F+1 -> bank (lane + i) % 64: conflict-free.
// ---------------------------------------------------------------------------
__global__ void __launch_bounds__(256)
kan_post_kernel(const float* __restrict__ Xred,   // (batch, IN_F)
                const float* __restrict__ Aff,    // (OUT_F, IN_F, 4)
                const int*   __restrict__ Fid,    // (OUT_F, IN_F)
                float*       __restrict__ Y,      // (batch, OUT_F)
                float*       __restrict__ Post)   // (batch, OUT_F, IN_F)
{
    __shared__ float xs[BT][IN_F + 1];
    __shared__ float ps[BT][IN_F + 1];
    __shared__ float ypart[8][BT];

    const int tid  = threadIdx.x;
    const int lane = tid & 31;
    const int b0   = blockIdx.x * BT;
    const int r    = blockIdx.y;

    __builtin_prefetch(Aff + (size_t)r * IN_F * 4, 0, 1);  // global_prefetch_b8

    // ---- stage x_red tile: coalesced float4 (wave covers one 512B row) ----
    {
        const int i0 = lane << 2;         // 0..124
        const int bb = tid >> 5;          // 0..7
#pragma unroll
        for (int p = 0; p < 4; ++p) {
            const int b = bb + p * 8;
            float4 v = *(const float4*)(Xred + (size_t)(b0 + b) * IN_F + i0);
            xs[b][i0 + 0] = v.x; xs[b][i0 + 1] = v.y;
            xs[b][i0 + 2] = v.z; xs[b][i0 + 3] = v.w;
        }
    }
    __syncthreads();

    // ---- compute: wave-uniform (r, i), lanes = batch ----
    const int wv = __builtin_amdgcn_readfirstlane(tid >> 5);  // wave id -> SGPR
    float partial = 0.0f;

    for (int t = 0; t < 16; ++t) {
        const int i   = wv * 16 + t;                               // uniform
        const int fid = __builtin_amdgcn_readfirstlane(Fid[r * IN_F + i]);
        const float4 abcd = *(const float4*)(Aff + ((size_t)r * IN_F + i) * 4);

        const float z = abcd.x * xs[lane][i] + abcd.y;   // a*x + b
        const float p = abcd.z * apply_fun_u(z, fid) + abcd.w;
        ps[lane][i] = p;
        partial += p;
    }
    ypart[wv][lane] = partial;
    __syncthreads();

    // ---- writeback postacts tile: coalesced float4 ----
    {
        const int i0 = lane << 2;
        const int bb = tid >> 5;
#pragma unroll
        for (int p = 0; p < 4; ++p) {
            const int b = bb + p * 8;
            float4 v = make_float4(ps[b][i0 + 0], ps[b][i0 + 1],
                                   ps[b][i0 + 2], ps[b][i0 + 3]);
            *(float4*)(Post + ((size_t)(b0 + b) * OUT_F + r) * IN_F + i0) = v;
        }
    }

    // ---- y[b, r] = sum over all i (combine 8 wave partials) ----
    if (tid < BT) {
        float s = 0.0f;
#pragma unroll
        for (int w = 0; w < 8; ++w) s += ypart[w][tid];
        Y[(size_t)(b0 + tid) * OUT_F + r] = s;
    }
}

// ---------------------------------------------------------------------------
extern "C" void kernel_launch(void* const* d_in, const int* in_sizes, int n_in,
                              void* d_out, int out_size, void* d_ws, size_t ws_size,
                              hipStream_t stream) {
    const float* x       = (const float*)d_in[0];  // (batch, 128)
    const float* affine  = (const float*)d_in[1];  // (128, 128, 4)
    const float* mask    = (const float*)d_in[2];  // (128, 128)
    const int*   fun_ids = (const int*)d_in[3];    // (128, 128)

    const int batch = in_sizes[0] / IN_F;          // 4096

    float* y    = (float*)d_out;                   // (batch, 128)
    float* post = y + (size_t)batch * OUT_F;       // (batch, 128, 128)
    float* xred = (float*)d_ws;                    // (batch, 128) scratch, 2 MB

    // 1) x_red = x @ mask^T via f32 WMMA
    dim3 g1(batch / 16, OUT_F / 16);
    xred_wmma_kernel<<<g1, 32, 0, stream>>>(x, mask, xred);

    // 2) divergence-free activation pass: (batch/32) x OUT_F workgroups
    dim3 g2(batch / BT, OUT_F);
    kan_post_kernel<<<g2, 256, 0, stream>>>(xred, affine, fun_ids, y, post);
}